// EdgeRotation_49435073577214
// MI455X (gfx1250) — compile-verified
//
#include <hip/hip_runtime.h>

typedef __attribute__((ext_vector_type(2))) float v2f;
typedef __attribute__((ext_vector_type(8))) float v8f;

#define WSW_FLOATS (16 * 16 * 32 * 2) /* 16384 floats = 64 KB, B-operand swizzled W */
#define SB_FLOATS  (8 * 16 * 16)      /* 8 waves x 16 edges x (a[8],b[8]) */
#define EPS_F 1e-7f

// Swizzled GEMM weights W[64][256] laid out per-lane for the WMMA B operand.
__device__ __align__(16) float g_Wsw[WSW_FLOATS];

__device__ __forceinline__ int iabs_(int v) { return v < 0 ? -v : v; }

// ---------------------------------------------------------------------------
// Setup: build W[64][256] (f64 math) where Out[e][ri*16+cj] = sum_k u[e][k]*W[k][...]
// u[k=p*8+q] = a[p]*b[q], a=(1,cos a,cos2a,cos3a,sin a,sin2a,sin3a,0), b same in beta.
// D^l = Za * (J * Zb * J); Za/Zb are diag+antidiag sparse -> coefficients below.
// Stored pre-swizzled: lane<16 holds K rows (4kc+0,4kc+1), lane>=16 rows (4kc+2,4kc+3),
// column n = lane&15, at float index ((kc*16 + jc)*32 + lane)*2 + d.
// ---------------------------------------------------------------------------
__global__ __launch_bounds__(256) void wigner_setup_kernel() {
  int t = threadIdx.x;          // one thread per output column (ri, cj)
  int ri = t >> 4, cj = t & 15;
  int ell, off;
  if (ri == 0)      { ell = 0; off = 0; }
  else if (ri <= 3) { ell = 1; off = 1; }
  else if (ri <= 8) { ell = 2; off = 4; }
  else              { ell = 3; off = 9; }
  int dim = 2 * ell + 1;

  double w[64];
#pragma unroll
  for (int k = 0; k < 64; ++k) w[k] = 0.0;

  if (cj >= off && cj < off + dim) {
    // ---- J^l = diag((-1)^i) * d^l(pi/2); note c=s=sqrt(1/2) so c^x s^y = 2^-l ----
    double fact[8];
    fact[0] = 1.0;
    for (int i2 = 1; i2 < 8; ++i2) fact[i2] = fact[i2 - 1] * (double)i2;
    double scale = 1.0;
    for (int i2 = 0; i2 < ell; ++i2) scale *= 0.5;

    double J[7][7];
    for (int mi = 0; mi < dim; ++mi) {
      int m = ell - mi;
      for (int mpi = 0; mpi < dim; ++mpi) {
        int mp = ell - mpi;
        double pref = sqrt(fact[ell + m] * fact[ell - m] * fact[ell + mp] * fact[ell - mp]);
        int k0 = (m - mp) > 0 ? (m - mp) : 0;
        int k1 = (ell + m) < (ell - mp) ? (ell + m) : (ell - mp);
        double val = 0.0;
        for (int k = k0; k <= k1; ++k) {
          double denom = fact[ell + m - k] * fact[ell - mp - k] * fact[k + mp - m] * fact[k];
          double sgn = ((m - mp + k) & 1) ? -1.0 : 1.0;
          val += sgn * pref / denom * scale;
        }
        J[mi][mpi] = ((mi & 1) ? -1.0 : 1.0) * val;
      }
    }

    // ---- coefficients of a_p * b_q in D[i][j] ----
    int i = ri - off, j = cj - off;
    int rev_i = dim - 1 - i;
    int m_i = ell - i;
    for (int term = 0; term < 2; ++term) {
      int p, r; double f;
      if (term == 0) { p = iabs_(m_i); f = 1.0; r = i; }             // cos(m_i a) * M[i][j]
      else {
        if (m_i == 0) break;                                        // sin(m_i a) * M[rev_i][j]
        p = 3 + iabs_(m_i); f = (m_i > 0) ? 1.0 : -1.0; r = rev_i;
      }
      for (int k = 0; k < dim; ++k) {
        int m_k = ell - k, rev_k = dim - 1 - k;
        // M[r][j] = sum_k J[r][k] * ( cos(m_k b) J[k][j] + sin(m_k b) J[rev_k][j] )
        w[p * 8 + iabs_(m_k)] += f * J[r][k] * J[k][j];
        if (m_k != 0)
          w[p * 8 + 3 + iabs_(m_k)] += f * ((m_k > 0) ? 1.0 : -1.0) * J[r][k] * J[rev_k][j];
      }
    }
  }

  // scatter into B-operand swizzled layout (jc = ri, n = cj); zero cols write zeros
#pragma unroll
  for (int k = 0; k < 64; ++k) {
    int kc = k >> 2, dlt = k & 3;
    int lane = cj + ((dlt >= 2) ? 16 : 0);
    int dd = dlt & 1;
    g_Wsw[(((kc * 16) + ri) * 32 + lane) * 2 + dd] = (float)w[k];
  }
}

// per output row jc: bitmask of K-chunks (of 4) where W has any nonzero
__constant__ constexpr unsigned kcMaskTab[16] = {
    0x0001u,                                            // l=0
    0x030Fu, 0x030Fu, 0x030Fu,                          // l=1
    0x0F3Fu, 0x0F3Fu, 0x0F3Fu, 0x0F3Fu, 0x0F3Fu,       // l=2
    0x3FFFu, 0x3FFFu, 0x3FFFu, 0x3FFFu, 0x3FFFu, 0x3FFFu, 0x3FFFu  // l=3
};

// ---------------------------------------------------------------------------
// Main: Out[E,256] = U[E,64] @ W[64,256] via V_WMMA_F32_16X16X4_F32.
// Fully unrolled: 167 WMMAs per 16-edge tile (zero K-chunks skipped at compile time).
// ---------------------------------------------------------------------------
__global__ __launch_bounds__(256) void wigner_main_kernel(
    const float* __restrict__ ev, float* __restrict__ out, int E) {
  extern __shared__ float smem[];
  float* sW = smem;               // 16384 floats (swizzled W)
  float* sB = smem + WSW_FLOATS;  // 2048 floats (per-wave trig basis)

  int tid = threadIdx.x;

  // stage swizzled W into LDS with CDNA5 async copy (ASYNCcnt-tracked, no VGPR data)
  {
    typedef __attribute__((address_space(3))) float lds_float_t;
    unsigned ldsBase = (unsigned)(unsigned long long)(lds_float_t*)sW;
    const char* gsrc = (const char*)g_Wsw;
#pragma unroll
    for (int it = 0; it < 16; ++it) {
      int idx = tid + it * 256;  // float4 index; 256 threads x 16 iters = 64 KB
      unsigned loff = ldsBase + (unsigned)idx * 16u;
      unsigned long long ga = (unsigned long long)(gsrc + (size_t)idx * 16u);
      asm volatile("global_load_async_to_lds_b128 %0, %1, off"
                   :: "v"(loff), "v"(ga) : "memory");
    }
    asm volatile("s_wait_asynccnt 0x0" ::: "memory");
  }
  __syncthreads();

  int wave = tid >> 5;
  int lane = tid & 31;
  long tile  = (long)blockIdx.x * 8 + wave;
  long eBase = tile * 16;
  float* bas = sB + wave * 256;   // 16 edges x 16 floats: a[0..7], b[0..7]

  if (lane < 16) {
    long e = eBase + lane;
    float x = 0.f, y = 0.f, z = 1.f;
    if (e < (long)E) { x = ev[e * 3 + 0]; y = ev[e * 3 + 1]; z = ev[e * 3 + 2]; }
    float r  = sqrtf(x * x + y * y + z * z);
    float cb = z / fmaxf(r, EPS_F);
    cb = fminf(fmaxf(cb, -1.f + EPS_F), 1.f - EPS_F);
    float sb = sqrtf(fmaxf(1.f - cb * cb, 0.f));        // beta in (0,pi) -> sin >= 0
    float ca, sa;
    if (fabsf(x) < EPS_F && fabsf(y) < EPS_F) { ca = 1.f; sa = 0.f; }  // alpha = 0
    else { float h = sqrtf(x * x + y * y); ca = x / h; sa = y / h; }   // atan2(y,x)
    float c2a = ca * ca - sa * sa, s2a = 2.f * ca * sa;
    float c3a = c2a * ca - s2a * sa, s3a = s2a * ca + c2a * sa;
    float c2b = cb * cb - sb * sb, s2b = 2.f * cb * sb;
    float c3b = c2b * cb - s2b * sb, s3b = s2b * cb + c2b * sb;
    float* bp = bas + lane * 16;
    bp[0] = 1.f; bp[1] = ca;  bp[2] = c2a; bp[3] = c3a;
    bp[4] = sa;  bp[5] = s2a; bp[6] = s3a; bp[7] = 0.f;
    bp[8] = 1.f; bp[9] = cb;  bp[10] = c2b; bp[11] = c3b;
    bp[12] = sb; bp[13] = s2b; bp[14] = s3b; bp[15] = 0.f;
  }
  __syncthreads();

  int half = lane >> 4;   // A/B operand K-row group
  int n    = lane & 15;   // A: edge row; B/D: output column within tile

  // Preload all A fragments once per tile (jc-independent): Af[kc] = (u[k0], u[k0+1])
  v2f Af[14];
#pragma unroll
  for (int kc = 0; kc < 14; ++kc) {
    int k0 = 4 * kc + 2 * half;
    int p = k0 >> 3, q = k0 & 7;           // q even -> q,q+1 share p, 8B-aligned pair
    float ap  = bas[n * 16 + p];
    float bq0 = bas[n * 16 + 8 + q];
    float bq1 = bas[n * 16 + 8 + q + 1];
    Af[kc].x = ap * bq0;
    Af[kc].y = ap * bq1;
  }

  const bool full = (eBase + 16 <= (long)E);             // wave-uniform
  float* obase = out + (size_t)eBase * 256 + (size_t)half * (8 * 256) + n;

#pragma unroll
  for (int jc = 0; jc < 16; ++jc) {          // jc = output row ri of the 16x16 D
    v8f acc = {0.f, 0.f, 0.f, 0.f, 0.f, 0.f, 0.f, 0.f};
#pragma unroll
    for (int kc = 0; kc < 14; ++kc) {
      if (kcMaskTab[jc] & (1u << kc)) {      // folds at compile time (both unrolled)
        v2f B = *(const v2f*)(sW + (((kc * 16) + jc) * 32 + lane) * 2);
        acc = __builtin_amdgcn_wmma_f32_16x16x4_f32(
            false, Af[kc], false, B, (short)0, acc, false, false);
      }
    }
    if (full) {
      // D layout: VGPR v -> edge row v + 8*half, column n; one addr pair + imm offsets
#pragma unroll
      for (int v = 0; v < 8; ++v)
        obase[(size_t)v * 256 + jc * 16] = acc[v];
    } else {
#pragma unroll
      for (int v = 0; v < 8; ++v) {
        long e = eBase + v + 8 * half;
        if (e < (long)E) out[e * 256 + (long)(jc * 16 + n)] = acc[v];
      }
    }
  }
}

extern "C" void kernel_launch(void* const* d_in, const int* in_sizes, int n_in,
                              void* d_out, int out_size, void* d_ws, size_t ws_size,
                              hipStream_t stream) {
  (void)n_in; (void)out_size; (void)d_ws; (void)ws_size;
  const float* ev = (const float*)d_in[0];
  float* out = (float*)d_out;
  int E = in_sizes[0] / 3;

  wigner_setup_kernel<<<1, 256, 0, stream>>>();

  size_t shbytes = (size_t)(WSW_FLOATS + SB_FLOATS) * sizeof(float);  // 72 KB
  (void)hipFuncSetAttribute((const void*)wigner_main_kernel,
                            hipFuncAttributeMaxDynamicSharedMemorySize, (int)shbytes);

  int nblocks = (E + 127) / 128;  // 8 waves/block x 16 edges/wave
  wigner_main_kernel<<<nblocks, 256, shbytes, stream>>>(ev, out, E);
}